// AmrstrongFrederickViscoplasticity_85298050498890
// MI455X (gfx1250) — compile-verified
//
#include <hip/hip_runtime.h>
#include <math.h>

// ---------------------------------------------------------------------------
// Armstrong–Frederick viscoplastic return mapping, radial-return reduction.
//
// The reference's 19x19 Newton system is reduced analytically:
//   depsp = dp * n,   da_k = dp*(n - g_k*a_k_old)/(1 + dp*g_k)
//   dev(sig_eff) || dev(B(dp)),  seq(dp) = vm(B(dp)) - 1.5*h(dp)
//   h(dp) = 2*mu*dp + (2/3) * sum_k C_k * dp/(1+dp*g_k)
//   B(dp) = sig_trial - (2/3) * sum_k C_k * a_k_old/(1+dp*g_k)
// leaving a scalar Newton on  g(dp) = dp - dt*(max(f,0)/KN)^3  with exact
// dual-number derivative.  Same fixed point as the reference Newton.
//
// The final stiffness application sig = sig_old + C6 * (deps - dp*n) is done
// with V_WMMA_F32_16X16X4_F32 (16 points x 6 comps, K=6 split 4+2), staging
// vectors through LDS per the CDNA5 A/B/D fragment layouts.
// ---------------------------------------------------------------------------

typedef __attribute__((ext_vector_type(2))) float v2f;
typedef __attribute__((ext_vector_type(8))) float v8f;

__device__ __constant__ float kNothing = 0.f; // (keep TU non-trivial)

#define E_MOD   200000.0f
#define NU_P    0.3f
#define LAM_C   (E_MOD * NU_P / ((1.0f + NU_P) * (1.0f - 2.0f * NU_P)))
#define MU_C    (E_MOD / (2.0f * (1.0f + NU_P)))
#define SIG0_C  300.0f
#define QV_C    150.0f
#define BV_C    10.0f
#define KN_C    100.0f
#define C1_C    40000.0f
#define C2_C    10000.0f
#define G1_C    400.0f
#define G2_C    100.0f
#define NEWTON_ITERS 25

// Scalar radial-return Newton.  Returns dp and the flow direction n[6].
__device__ __forceinline__ void solve_point(const float S[6], const float a1[6],
                                            const float a2[6], float p0, float dt,
                                            float& dp_out, float n_out[6]) {
    const float invKN = 1.0f / KN_C;
    float dp = 0.0f;
#pragma unroll 1
    for (int it = 0; it < NEWTON_ITERS; ++it) {
        // dual numbers: value + d/d(dp)
        float den1 = 1.0f + dp * G1_C, den2 = 1.0f + dp * G2_C;
        float w1 = (2.0f / 3.0f) * C1_C / den1;
        float w2 = (2.0f / 3.0f) * C2_C / den2;
        float w1d = -w1 * G1_C / den1;
        float w2d = -w2 * G2_C / den2;
        float h  = 2.0f * MU_C * dp + dp * (w1 + w2);
        float hd = 2.0f * MU_C + (w1 + w2) + dp * (w1d + w2d);

        float B[6], Bd[6];
#pragma unroll
        for (int c = 0; c < 6; ++c) {
            B[c]  = S[c] - w1 * a1[c] - w2 * a2[c];
            Bd[c] = -w1d * a1[c] - w2d * a2[c];
        }
        float trB  = (B[0] + B[1] + B[2]) * (1.0f / 3.0f);
        float trBd = (Bd[0] + Bd[1] + Bd[2]) * (1.0f / 3.0f);
        float dd = 0.0f, ddd = 0.0f;
#pragma unroll
        for (int c = 0; c < 6; ++c) {
            float db  = B[c]  - (c < 3 ? trB  : 0.0f);
            float dbd = Bd[c] - (c < 3 ? trBd : 0.0f);
            dd  = fmaf(db, db, dd);
            ddd = fmaf(db, dbd, ddd);
        }
        float vm2 = fmaxf(1.5f * dd, 1e-16f);
        float vm  = sqrtf(vm2);
        float vmd = (1.5f * dd > 1e-16f) ? (1.5f * ddd / vm) : 0.0f;
        float seq  = vm - 1.5f * h;
        float seqd = vmd - 1.5f * hd;
        float pe = __expf(-BV_C * (p0 + dp));
        float f  = seq - (SIG0_C + QV_C * (1.0f - pe));
        float fd = seqd - QV_C * BV_C * pe;
        float fp = fmaxf(f, 0.0f) * invKN;
        float phi  = dt * fp * fp * fp;
        float phid = (f > 0.0f) ? (dt * 3.0f * fp * fp * fd * invKN) : 0.0f;
        float g  = dp - phi;
        float gd = 1.0f - phid;
        gd = (fabsf(gd) > 1e-12f) ? gd : 1e-12f;
        dp = fmaxf(dp - g / gd, 0.0f);
    }
    // final flow direction: n = 1.5*dev(B)/vm(B)
    float den1 = 1.0f + dp * G1_C, den2 = 1.0f + dp * G2_C;
    float w1 = (2.0f / 3.0f) * C1_C / den1;
    float w2 = (2.0f / 3.0f) * C2_C / den2;
    float B[6];
#pragma unroll
    for (int c = 0; c < 6; ++c) B[c] = S[c] - w1 * a1[c] - w2 * a2[c];
    float trB = (B[0] + B[1] + B[2]) * (1.0f / 3.0f);
    float dd = 0.0f;
    float devB[6];
#pragma unroll
    for (int c = 0; c < 6; ++c) {
        devB[c] = B[c] - (c < 3 ? trB : 0.0f);
        dd = fmaf(devB[c], devB[c], dd);
    }
    float vm = sqrtf(fmaxf(1.5f * dd, 1e-16f));
    float inv = 1.5f / vm;
#pragma unroll
    for (int c = 0; c < 6; ++c) n_out[c] = devB[c] * inv;
    dp_out = dp;
}

// Constant stiffness matrix entry C6[r][c] (symmetric): LAM on 3x3 block + 2MU*I.
__device__ __forceinline__ float c6_entry(int r, int c) {
    float v = 0.0f;
    v += (r < 3 && c < 3) ? LAM_C : 0.0f;
    v += (r == c && r < 6) ? (2.0f * MU_C) : 0.0f;
    v = (r < 6 && c < 6) ? v : 0.0f;
    return v;
}

// Main kernel: full 128-thread blocks only (EXEC all-1s guaranteed for WMMA).
__global__ __launch_bounds__(128)
void afvp_wmma_kernel(const float* __restrict__ eps, const float* __restrict__ eps_old,
                      const float* __restrict__ sig_old, const float* __restrict__ p_old,
                      const float* __restrict__ epsp_old, const float* __restrict__ a_old,
                      const float* __restrict__ dtp, float* __restrict__ out, int N) {
    __shared__ float lds[128 * 8];   // per-wave 32 points x 8 floats (padded 6-vec)

    const int tid  = threadIdx.x;
    const int idx  = blockIdx.x * 128 + tid;
    const int lane = tid & 31;
    const int wbase = (tid & ~31) * 8;     // this wave's LDS base (floats)
    const float dt = dtp[0];

    // ---- load point state -------------------------------------------------
    float deps[6], S[6], a1[6], a2[6], sg_o[6];
#pragma unroll
    for (int c = 0; c < 6; ++c) {
        float e  = eps[idx * 6 + c];
        float eo = eps_old[idx * 6 + c];
        deps[c] = e - eo;
        sg_o[c] = sig_old[idx * 6 + c];
        a1[c] = a_old[idx * 12 + c];
        a2[c] = a_old[idx * 12 + 6 + c];
    }
    float trd = deps[0] + deps[1] + deps[2];
#pragma unroll
    for (int c = 0; c < 6; ++c)
        S[c] = sg_o[c] + (c < 3 ? LAM_C * trd : 0.0f) + 2.0f * MU_C * deps[c];
    const float p0 = p_old[idx];

    // ---- scalar Newton (radial return) ------------------------------------
    float dp, n[6];
    solve_point(S, a1, a2, p0, dt, dp, n);

    // ---- stage x = deps - dp*n into LDS for WMMA A fragments ---------------
#pragma unroll
    for (int c = 0; c < 6; ++c)
        lds[wbase + lane * 8 + c] = fmaf(-dp, n[c], deps[c]);
    __syncthreads();

    // ---- C6 * x via V_WMMA_F32_16X16X4_F32 (two tiles of 16 points) --------
    // A 16x4 layout: lanes 0-15 = rows M, VGPR0/1 = K{0,1}; lanes 16-31 = K{2,3}.
    // B 4x16 layout: VGPR r holds K=r (lanes 0-15) / K=r+2 (lanes 16-31), N=lane&15.
    // D 16x16 layout: VGPR r, lane l -> M = r + 8*(l>=16), Ncol = l&15.
    const int m  = lane & 15;
    const int kh = lane >> 4;
    v2f b1, b2;
    b1.x = c6_entry(m, 2 * kh + 0);
    b1.y = c6_entry(m, 2 * kh + 1);
    b2.x = c6_entry(m, 2 * kh + 4);
    b2.y = c6_entry(m, 2 * kh + 5);

    v8f dacc[2];
#pragma unroll
    for (int t = 0; t < 2; ++t) {
        const float* X = &lds[wbase + t * 16 * 8];
        v2f aA, aB;
        aA.x = X[m * 8 + 2 * kh + 0];
        aA.y = X[m * 8 + 2 * kh + 1];
        aB.x = kh ? 0.0f : X[m * 8 + 4];     // K=4..5 in half 0, zero pad half 1
        aB.y = kh ? 0.0f : X[m * 8 + 5];
        v8f acc = {};
        acc = __builtin_amdgcn_wmma_f32_16x16x4_f32(false, aA, false, b1,
                                                    (short)0, acc, false, false);
        acc = __builtin_amdgcn_wmma_f32_16x16x4_f32(false, aB, false, b2,
                                                    (short)0, acc, false, false);
        dacc[t] = acc;
    }
    __syncthreads();

    // scatter D back: lane holds column (l&15) for rows r + 8*kh of each tile
    if (m < 6) {
#pragma unroll
        for (int t = 0; t < 2; ++t)
#pragma unroll
            for (int r = 0; r < 8; ++r) {
                int row = t * 16 + r + 8 * kh;
                lds[wbase + row * 8 + m] = dacc[t][r];
            }
    }
    __syncthreads();

    // ---- outputs: sig | p | epsp | a  (flat concat, reference return order) -
    const int NN = N;
    float den1 = 1.0f + dp * G1_C, den2 = 1.0f + dp * G2_C;
#pragma unroll
    for (int c = 0; c < 6; ++c) {
        float csig = lds[wbase + lane * 8 + c];          // C6 * (deps - dp*n)
        out[idx * 6 + c] = sg_o[c] + csig;               // sig
        float depsp = dp * n[c];
        out[7 * NN + idx * 6 + c] = epsp_old[idx * 6 + c] + depsp;   // epsp
        float da1 = dp * (n[c] - G1_C * a1[c]) / den1;
        float da2 = dp * (n[c] - G2_C * a2[c]) / den2;
        out[13 * NN + idx * 12 + c]     = a1[c] + da1;   // a[:,0,:]
        out[13 * NN + idx * 12 + 6 + c] = a2[c] + da2;   // a[:,1,:]
    }
    out[6 * NN + idx] = p0 + dp;                          // p
}

// Scalar tail kernel for N % 128 points (no WMMA, EXEC-safe with bounds check).
__global__ __launch_bounds__(128)
void afvp_tail_kernel(const float* __restrict__ eps, const float* __restrict__ eps_old,
                      const float* __restrict__ sig_old, const float* __restrict__ p_old,
                      const float* __restrict__ epsp_old, const float* __restrict__ a_old,
                      const float* __restrict__ dtp, float* __restrict__ out,
                      int start, int N) {
    int idx = start + blockIdx.x * 128 + threadIdx.x;
    if (idx >= N) return;
    const float dt = dtp[0];
    float deps[6], S[6], a1[6], a2[6], sg_o[6];
#pragma unroll
    for (int c = 0; c < 6; ++c) {
        deps[c] = eps[idx * 6 + c] - eps_old[idx * 6 + c];
        sg_o[c] = sig_old[idx * 6 + c];
        a1[c] = a_old[idx * 12 + c];
        a2[c] = a_old[idx * 12 + 6 + c];
    }
    float trd = deps[0] + deps[1] + deps[2];
#pragma unroll
    for (int c = 0; c < 6; ++c)
        S[c] = sg_o[c] + (c < 3 ? LAM_C * trd : 0.0f) + 2.0f * MU_C * deps[c];
    float p0 = p_old[idx];
    float dp, n[6];
    solve_point(S, a1, a2, p0, dt, dp, n);
    float den1 = 1.0f + dp * G1_C, den2 = 1.0f + dp * G2_C;
#pragma unroll
    for (int c = 0; c < 6; ++c) {
        out[idx * 6 + c] = S[c] - 2.0f * MU_C * dp * n[c];
        out[7 * N + idx * 6 + c] = epsp_old[idx * 6 + c] + dp * n[c];
        out[13 * N + idx * 12 + c]     = a1[c] + dp * (n[c] - G1_C * a1[c]) / den1;
        out[13 * N + idx * 12 + 6 + c] = a2[c] + dp * (n[c] - G2_C * a2[c]) / den2;
    }
    out[6 * N + idx] = p0 + dp;
}

extern "C" void kernel_launch(void* const* d_in, const int* in_sizes, int n_in,
                              void* d_out, int out_size, void* d_ws, size_t ws_size,
                              hipStream_t stream) {
    const float* eps      = (const float*)d_in[0];
    const float* eps_old  = (const float*)d_in[1];
    const float* sig_old  = (const float*)d_in[2];
    const float* p_old    = (const float*)d_in[3];
    const float* epsp_old = (const float*)d_in[4];
    const float* a_old    = (const float*)d_in[5];
    const float* dtp      = (const float*)d_in[6];
    float* out = (float*)d_out;
    const int N = in_sizes[3];           // p_old has N elements

    const int nfull = N / 128;
    if (nfull > 0) {
        afvp_wmma_kernel<<<nfull, 128, 0, stream>>>(eps, eps_old, sig_old, p_old,
                                                    epsp_old, a_old, dtp, out, N);
    }
    const int start = nfull * 128;
    const int rem = N - start;
    if (rem > 0) {
        afvp_tail_kernel<<<(rem + 127) / 128, 128, 0, stream>>>(
            eps, eps_old, sig_old, p_old, epsp_old, a_old, dtp, out, start, N);
    }
}